// CFDFVGCN_86122684219977
// MI455X (gfx1250) — compile-verified
//
#include <hip/hip_runtime.h>
#include <hip/hip_bf16.h>

#define N_NODES 50000
#define N_EDGES 800000
#define C_IN    64
#define KDIM    192   // HID * C_IN
#define C_OUT   128
#define ATTR    6
#define M_TILES (N_NODES / 16)   // 3125, exact
#define KHALF   96               // stage W_out in two 48KB halves

typedef __attribute__((ext_vector_type(2))) float v2f;
typedef __attribute__((ext_vector_type(8))) float v8f;

// ---------------------------------------------------------------------------
// Zero the aggregation workspace (both layers) — deterministic per call.
// ---------------------------------------------------------------------------
__global__ __launch_bounds__(256) void zero_kernel(float* __restrict__ p, size_t n) {
    size_t i = (size_t)blockIdx.x * blockDim.x + threadIdx.x;
    size_t stride = (size_t)gridDim.x * blockDim.x;
    for (; i < n; i += stride) p[i] = 0.0f;
}

// ---------------------------------------------------------------------------
// Edge stage: each wave processes 32 edges per chunk.
// Phase 1: lane L loads edge (base+L)'s src/dst/attr — fully coalesced.
//          Prefetch the gathered x rows (global_prefetch_b8).
// Phase 2: iterate the 32 edges; broadcast metadata via __shfl (lane ops,
//          not memory); lanes cover the 192 message channels (6 each):
//   scaling[k] = relu(b_in[k] + sum_a attr[a] * W_in[a][k])
//   msg[k]     = scaling[k] * x[src][k/3]
//   atomicAdd(aggr[dst][k], msg[k])   (coalesced; aggr is L2-resident)
// ---------------------------------------------------------------------------
__global__ __launch_bounds__(256) void edge_kernel(
    const float*     __restrict__ x,      // [N_NODES, C_IN]
    const long long* __restrict__ ei,     // [2, N_EDGES] int64
    const float*     __restrict__ ea,     // [N_EDGES, ATTR]
    const float*     __restrict__ W_in,   // [ATTR, KDIM]
    const float*     __restrict__ b_in,   // [KDIM]
    float*           __restrict__ aggr)   // [N_NODES, KDIM], pre-zeroed
{
    __shared__ float sW[ATTR * KDIM];
    __shared__ float sB[KDIM];
    for (int i = threadIdx.x; i < ATTR * KDIM; i += blockDim.x) sW[i] = W_in[i];
    for (int i = threadIdx.x; i < KDIM;        i += blockDim.x) sB[i] = b_in[i];
    __syncthreads();

    const int lane   = threadIdx.x & 31;
    const int wpb    = blockDim.x >> 5;
    const int waveId = blockIdx.x * wpb + (threadIdx.x >> 5);
    const int nWaves = gridDim.x * wpb;

    // N_EDGES % 32 == 0 -> every in-range chunk is full, no tail handling.
    for (int base = waveId * 32; base < N_EDGES; base += nWaves * 32) {
        const int eL   = base + lane;
        const int srcL = (int)ei[eL];
        const int dstL = (int)ei[N_EDGES + eL];
        float atL[ATTR];
#pragma unroll
        for (int a = 0; a < ATTR; ++a) atL[a] = ea[(size_t)eL * ATTR + a];

        // Pull this chunk's 32 gather rows toward the caches ahead of use.
        __builtin_prefetch(x + (size_t)srcL * C_IN, 0, 3);

        for (int q = 0; q < 32; ++q) {
            const int src = __shfl(srcL, q);
            const int dst = __shfl(dstL, q);
            const float a0 = __shfl(atL[0], q);
            const float a1 = __shfl(atL[1], q);
            const float a2 = __shfl(atL[2], q);
            const float a3 = __shfl(atL[3], q);
            const float a4 = __shfl(atL[4], q);
            const float a5 = __shfl(atL[5], q);

            const float* __restrict__ xr = x + (size_t)src * C_IN;
            float* __restrict__ ar = aggr + (size_t)dst * KDIM;

#pragma unroll
            for (int j = 0; j < 6; ++j) {
                const int k = lane + 32 * j;                  // 0..191
                float s = sB[k]
                        + a0 * sW[0 * KDIM + k]
                        + a1 * sW[1 * KDIM + k]
                        + a2 * sW[2 * KDIM + k]
                        + a3 * sW[3 * KDIM + k]
                        + a4 * sW[4 * KDIM + k]
                        + a5 * sW[5 * KDIM + k];
                s = fmaxf(s, 0.0f);                           // relu
                const float xv = xr[k / 3];                   // channel c = k/3
                atomicAdd(&ar[k], s * xv);                    // global_atomic_add_f32
            }
        }
    }
}

// ---------------------------------------------------------------------------
// Node stage: out[m, colOff+n] = tanh( sum_k aggr[m,k] * W_out[k,n] + b_out[n] )
// One wave per 16-row strip; 8 column tiles of 16; K stepped by 4 via
// V_WMMA_F32_16X16X4_F32. W_out is staged into LDS in two 48KB halves with
// the CDNA5 async global->LDS path (ASYNCcnt), shared by all 4 waves; B
// operands then come from LDS while the global pipe streams A (aggr rows).
//
// WMMA VGPR layouts (wave32):
//   A 16x4 f32 : lane L<16 -> {A[M=L,K=0], A[M=L,K=1]}, L>=16 -> K=2,3
//   B 4x16 f32 : lane L<16 -> {B[K=0,N=L], B[K=1,N=L]}, L>=16 -> K=2,3
//   C/D 16x16  : element r of v8f -> M = r + 8*(L>>4), N = L&15
// ---------------------------------------------------------------------------
__global__ __launch_bounds__(128) void node_gemm_kernel(
    const float* __restrict__ aggr,    // [N_NODES, KDIM]
    const float* __restrict__ W_out,   // [KDIM, C_OUT]
    const float* __restrict__ b_out,   // [C_OUT]
    float*       __restrict__ out,     // [N_NODES, 2*C_OUT]
    int colOffset)                     // 0 (layer 0) or 128 (layer 1)
{
    __shared__ float sW[KHALF * C_OUT];               // 48KB staging buffer

    const int wave  = threadIdx.x >> 5;
    const int lane  = threadIdx.x & 31;
    const int mTile = blockIdx.x * 4 + wave;          // wave-uniform
    const bool active = (mTile < M_TILES);            // tail waves stay for barriers
    const int m0 = (active ? mTile : (M_TILES - 1)) * 16;  // clamp: no OOB reads

    const int l15 = lane & 15;
    const int kg  = lane >> 4;                        // 0 or 1: K pair group

    v8f acc[8] = {};
    const float* __restrict__ arow = aggr + (size_t)(m0 + l15) * KDIM;
    // Generic pointer to LDS: low 32 bits are the LDS byte address.
    const unsigned ldsBase = (unsigned)(uintptr_t)(void*)&sW[0];

    for (int half = 0; half < 2; ++half) {
        __syncthreads();                              // previous half consumed
        // Stage KHALF rows of W_out (48KB) via async b128 copies:
        // 3072 chunks of 16B, 24 per thread. GVS: mem = SGPR(W_out)+VGPR off.
        const unsigned srcBase = (unsigned)(half * KHALF * C_OUT * (int)sizeof(float));
#pragma unroll
        for (int i = 0; i < (KHALF * C_OUT * 4) / (16 * 128); ++i) {   // 24
            const unsigned chunk   = ((unsigned)threadIdx.x + i * 128u) * 16u;
            const unsigned ldsAddr = ldsBase + chunk;
            const unsigned gOff    = srcBase + chunk;
            asm volatile("global_load_async_to_lds_b128 %0, %1, %2 offset:0"
                         :: "v"(ldsAddr), "v"(gOff), "s"(W_out) : "memory");
        }
        asm volatile("s_wait_asynccnt 0x0" ::: "memory");
        __syncthreads();                              // LDS visible to all waves

        for (int kk = 0; kk < KHALF; kk += 4) {
            const int k0 = half * KHALF + kk;
            v2f a;
            a.x = arow[k0 + 2 * kg + 0];
            a.y = arow[k0 + 2 * kg + 1];
#pragma unroll
            for (int t = 0; t < 8; ++t) {
                const int n0 = t * 16;
                v2f b;
                b.x = sW[(kk + 2 * kg + 0) * C_OUT + n0 + l15];
                b.y = sW[(kk + 2 * kg + 1) * C_OUT + n0 + l15];
                // 8 args: (neg_a, A, neg_b, B, c_mod, C, reuse_a, reuse_b)
                acc[t] = __builtin_amdgcn_wmma_f32_16x16x4_f32(
                    false, a, false, b, (short)0, acc[t], false, false);
            }
        }
    }

    if (active) {
#pragma unroll
        for (int t = 0; t < 8; ++t) {
            const int n = t * 16 + l15;
            const float bias = b_out[n];
#pragma unroll
            for (int r = 0; r < 8; ++r) {
                const int m = m0 + r + 8 * kg;
                out[(size_t)m * (2 * C_OUT) + colOffset + n] = tanhf(acc[t][r] + bias);
            }
        }
    }
}

// ---------------------------------------------------------------------------
// Launcher
// Inputs (setup_inputs order):
//  0:x 1:edge_index0(i64) 2:edge_attr0 3:edge_index1(i64) 4:edge_attr1
//  5:W_in0 6:b_in0 7:W_out0 8:b_out0 9:W_in1 10:b_in1 11:W_out1 12:b_out1
// Workspace: aggr0 [N,192] f32, then aggr1 [N,192] f32  (76.8 MB total)
// ---------------------------------------------------------------------------
extern "C" void kernel_launch(void* const* d_in, const int* in_sizes, int n_in,
                              void* d_out, int out_size, void* d_ws, size_t ws_size,
                              hipStream_t stream) {
    const float*     x      = (const float*)d_in[0];
    const long long* ei0    = (const long long*)d_in[1];
    const float*     ea0    = (const float*)d_in[2];
    const long long* ei1    = (const long long*)d_in[3];
    const float*     ea1    = (const float*)d_in[4];
    const float*     W_in0  = (const float*)d_in[5];
    const float*     b_in0  = (const float*)d_in[6];
    const float*     W_out0 = (const float*)d_in[7];
    const float*     b_out0 = (const float*)d_in[8];
    const float*     W_in1  = (const float*)d_in[9];
    const float*     b_in1  = (const float*)d_in[10];
    const float*     W_out1 = (const float*)d_in[11];
    const float*     b_out1 = (const float*)d_in[12];

    float* out   = (float*)d_out;
    float* aggr0 = (float*)d_ws;
    float* aggr1 = aggr0 + (size_t)N_NODES * KDIM;

    const size_t aggrElems = 2 * (size_t)N_NODES * KDIM;
    zero_kernel<<<2048, 256, 0, stream>>>(aggr0, aggrElems);

    edge_kernel<<<2048, 256, 0, stream>>>(x, ei0, ea0, W_in0, b_in0, aggr0);
    edge_kernel<<<2048, 256, 0, stream>>>(x, ei1, ea1, W_in1, b_in1, aggr1);

    const int gemmBlocks = (M_TILES + 3) / 4;         // 4 waves / 16-row strips
    node_gemm_kernel<<<gemmBlocks, 128, 0, stream>>>(aggr0, W_out0, b_out0, out, 0);
    node_gemm_kernel<<<gemmBlocks, 128, 0, stream>>>(aggr1, W_out1, b_out1, out, C_OUT);
}